// CREDITHEALPix_90975997264316
// MI455X (gfx1250) — compile-verified
//
#include <hip/hip_runtime.h>
#include <hip/hip_bf16.h>
#include <stdint.h>

// HEALPix pad P=1: pure data movement, ~410MB traffic -> ~18us @ 23.3TB/s.
// Single fused kernel: each single-wave block issues a TDM tensor_load_to_lds
// (128x32 f32 tile), gathers its 129 border-ring elements while the DMA runs,
// then tensor_store_from_lds writes the tile into the padded interior with
// tensor_dim0_stride = 130 (the TDM engine does the strided scatter).

#define FN    128   // face edge
#define FNP   130   // padded edge (n + 2*P)
#define TROWS 32    // rows per TDM tile (16KB LDS); 4 tiles per plane
#define ROWS  8     // rows per block in the (unused fallback) async kernel

typedef unsigned int v4u __attribute__((ext_vector_type(4)));
typedef unsigned int v8u __attribute__((ext_vector_type(8)));

// Neighbor tables per face (derived from _pn/_pe/_ps call sites in reference).
// Faces 0-3 north polar, 4-7 equatorial, 8-11 south polar.
__device__ __constant__ signed char c_T [12] = {1,2,3,0,  0,1,2,3,  5,6,7,4};
__device__ __constant__ signed char c_TL[12] = {2,3,0,1,  0,0,0,0,  0,1,2,3};   // eq: interpolated
__device__ __constant__ signed char c_L [12] = {3,0,1,2,  3,0,1,2,  4,5,6,7};
__device__ __constant__ signed char c_BL[12] = {3,0,1,2,  7,4,5,6, 11,8,9,10};
__device__ __constant__ signed char c_B [12] = {4,5,6,7, 11,8,9,10,11,8,9,10};
__device__ __constant__ signed char c_BR[12] = {8,9,10,11,0,0,0,0, 10,11,8,9};  // eq: interpolated
__device__ __constant__ signed char c_R [12] = {5,6,7,4,  8,9,10,11,9,10,11,8};
__device__ __constant__ signed char c_TR[12] = {1,2,3,0,  5,6,7,4,  9,10,11,8};

// ---------------------------------------------------------------------------
// Build a TDM descriptor (D# groups 0 and 1) for a 2D f32 tile (ISA 08 §8).
// ---------------------------------------------------------------------------
__device__ __forceinline__ void tdm_desc2d(
    v4u& g0, v8u& g1, unsigned lds_addr, unsigned long long gaddr_bytes,
    unsigned tensor_d0, unsigned tensor_d1,
    unsigned tile_d0, unsigned tile_d1, unsigned long long stride0) {
  g0.x = 1u;                                              // count=1 (valid, user mode)
  g0.y = lds_addr;                                        // LDS byte address
  g0.z = (unsigned)(gaddr_bytes & 0xFFFFFFFFu);           // global_addr[31:0]
  g0.w = ((unsigned)(gaddr_bytes >> 32) & 0x01FFFFFFu)    // global_addr[56:32]
       | (2u << 30);                                      // type = 2 ("image")
  g1.s0 = 2u << 16;                                       // data_size=2 (4 bytes); mask/pad/iterate=0
  g1.s1 = (tensor_d0 & 0xFFFFu) << 16;                    // [63:48]=tensor_dim0 lo ; barrier addr=0
  g1.s2 = (tensor_d0 >> 16) | ((tensor_d1 & 0xFFFFu) << 16); // dim0 hi | dim1 lo
  g1.s3 = (tensor_d1 >> 16) | ((tile_d0 & 0xFFFFu) << 16);   // dim1 hi | tile_dim0
  g1.s4 = tile_d1 & 0xFFFFu;                              // tile_dim1 ; tile_dim2=0 (2D)
  g1.s5 = (unsigned)(stride0 & 0xFFFFFFFFu);              // tensor_dim0_stride[31:0]
  g1.s6 = (unsigned)(stride0 >> 32) & 0xFFFFu;            // stride0[47:32] ; stride1=0
  g1.s7 = 0u;
}

// ---------------------------------------------------------------------------
// Fused kernel: TDM interior tile + 129 border-ring elements per block.
// grid = planes*4 blocks of 32 threads (one wave).
// ---------------------------------------------------------------------------
__global__ void __launch_bounds__(32) healpix_pad_fused(
    const float* __restrict__ x, float* __restrict__ out) {
  __shared__ float lbuf[TROWS * FN];   // 16 KB

  const int bid   = blockIdx.x;
  const int plane = bid >> 2;          // plane = b12*C + ch
  const int tile  = bid & 3;
  const int r0    = tile * TROWS;

  // ---- issue TDM load: 128x32 contiguous tile -> LDS -------------------
  const unsigned lds_addr = (unsigned)(uintptr_t)(&lbuf[0]);  // low 32b of flat = LDS offset
  const unsigned long long gsrc =
      (unsigned long long)(uintptr_t)(x + (size_t)plane * (FN * FN) + (size_t)r0 * FN);
  const unsigned long long gdst =
      (unsigned long long)(uintptr_t)(out + (size_t)plane * (FNP * FNP)
                                      + (size_t)(r0 + 1) * FNP + 1);

  v4u l0; v8u l1;
  tdm_desc2d(l0, l1, lds_addr, gsrc, FN, TROWS, FN, TROWS, FN);    // contiguous in
  v4u s0; v8u s1;
  tdm_desc2d(s0, s1, lds_addr, gdst, FN, TROWS, FN, TROWS, FNP);   // 130-stride out

  asm volatile("tensor_load_to_lds %0, %1" :: "s"(l0), "s"(l1) : "memory");

  // ---- border ring, overlapped with the DMA ----------------------------
  // 516 ring elements per plane = 4 tiles * 129; this wave does its 129.
  {
    const int ch    = plane % 128;
    const int b12   = plane / 128;
    const int face  = b12 % 12;
    const int batch = b12 / 12;
    const int typ   = face >> 2;        // 0=north, 1=equatorial, 2=south
    const int NN    = FN - 1;           // 127
    const int lane  = threadIdx.x;      // 0..31

    auto LD = [&](int sface, int i, int j) -> float {
      size_t idx = ((((size_t)(batch * 12 + sface)) * 128 + ch) * FN + i) * FN + j;
      return x[idx];
    };

#pragma unroll
    for (int k = 0; k < 5; ++k) {
      const int e = k * 32 + lane;      // 0..128 valid (129 per tile)
      if (e < 129) {
        const int t = tile * 129 + e;   // ring index 0..515
        int r, c;
        if      (t < 130) { r = 0;           c = t;           }
        else if (t < 260) { r = FNP - 1;     c = t - 130;     }
        else if (t < 388) { r = t - 260 + 1; c = 0;           }
        else              { r = t - 388 + 1; c = FNP - 1;     }

        float v;
        if (r == 0 && c == 0) {                                   // top-left
          if      (typ == 0) v = LD(c_TL[face], 0, 0);
          else if (typ == 2) v = LD(c_TL[face], NN, NN);
          else               v = 0.5f * (LD(c_T[face], NN, 0) + LD(c_L[face], 0, NN));
        } else if (r == 0 && c == FNP - 1) {                      // top-right
          v = LD(c_TR[face], NN, 0);
        } else if (r == FNP - 1 && c == 0) {                      // bottom-left
          v = LD(c_BL[face], 0, NN);
        } else if (r == FNP - 1 && c == FNP - 1) {                // bottom-right
          if      (typ == 0) v = LD(c_BR[face], 0, 0);
          else if (typ == 2) v = LD(c_BR[face], NN, NN);
          else               v = 0.5f * (LD(c_B[face], 0, NN) + LD(c_R[face], NN, 0));
        } else if (r == 0) {                                      // top edge
          int j = c - 1;
          v = (typ == 0) ? LD(c_T[face], j, 0) : LD(c_T[face], NN, j);
        } else if (r == FNP - 1) {                                // bottom edge
          int j = c - 1;
          v = (typ == 2) ? LD(c_B[face], j, NN) : LD(c_B[face], 0, j);
        } else if (c == 0) {                                      // left edge
          int i = r - 1;
          v = (typ == 0) ? LD(c_L[face], 0, i) : LD(c_L[face], i, NN);
        } else {                                                  // right edge
          int i = r - 1;
          v = (typ == 2) ? LD(c_R[face], NN, i) : LD(c_R[face], i, 0);
        }
        out[((size_t)plane * FNP + r) * FNP + c] = v;
      }
    }
  }

  // ---- drain DMA, then TDM store: LDS -> padded interior (stride 130) --
  asm volatile("s_wait_tensorcnt 0x0" ::: "memory");
  asm volatile("tensor_store_from_lds %0, %1" :: "s"(s0), "s"(s1) : "memory");
  // S_ENDPGM performs an implicit wait-idle (TENSORcnt drains before exit).
}

// ---------------------------------------------------------------------------
// Fallback interior using the async global->LDS pipeline (ASYNCcnt).
// Not launched; kept as the proven alternative data-movement path.
// ---------------------------------------------------------------------------
__global__ void __launch_bounds__(256) healpix_interior_async(
    const float* __restrict__ x, float* __restrict__ out) {
  __shared__ float lbuf[ROWS * FN];

  const int bid   = blockIdx.x;
  const int plane = bid >> 4;
  const int r0    = (bid & 15) * ROWS;
  const int tid   = threadIdx.x;
  const int wave  = tid >> 5;
  const int lane  = tid & 31;

  const float* src = x + (size_t)plane * (FN * FN) + (size_t)(r0 + wave) * FN + lane * 4;
  unsigned lds_off = (unsigned)(uintptr_t)(&lbuf[wave * FN + lane * 4]);
  asm volatile("global_load_async_to_lds_b128 %0, %1, off"
               :: "v"(lds_off), "v"(src) : "memory");
  asm volatile("s_wait_asynccnt 0x0" ::: "memory");
  __syncthreads();

  float* ob = out + (size_t)plane * (FNP * FNP) + (size_t)(r0 + 1) * FNP + 1;
#pragma unroll
  for (int k = 0; k < 4; ++k) {
    int e  = tid + k * 256;
    int rr = e >> 7;
    int cc = e & (FN - 1);
    ob[rr * FNP + cc] = lbuf[e];
  }
}

// ---------------------------------------------------------------------------
extern "C" void kernel_launch(void* const* d_in, const int* in_sizes, int n_in,
                              void* d_out, int out_size, void* d_ws, size_t ws_size,
                              hipStream_t stream) {
  const float* x   = (const float*)d_in[0];
  float*       out = (float*)d_out;

  const int planes = in_sizes[0] / (FN * FN);   // 24 * 128 = 3072

  // one launch: TDM interior tile + overlapped border gather per block
  healpix_pad_fused<<<planes * 4, 32, 0, stream>>>(x, out);
}